// ACE15TEModel_41824391528638
// MI455X (gfx1250) — compile-verified
//
#include <hip/hip_runtime.h>
#include <hip/hip_bf16.h>
#include <stdint.h>

#define T_STEPS     512
#define VOCAB       155776
#define AUDIO_START 151669
#define EOS_TOK     151645
#define NCAND       (VOCAB - AUDIO_START + 1)   // 4107 audio tokens + EOS slot = 4108
#define NAUDIO      (VOCAB - AUDIO_START)       // 4107
#define TOPK        64
#define TOPP        0.9f
#define TEMP        0.85f
#define NEGF        (-3.4028234663852886e38f)   // finfo(float32).min
#define TINYF       (1.17549435e-38f)
#define NTHREADS    256

#define NQUAD       (AUDIO_START / 4)           // 37917 float4 quads in [0, 151668)
#define TAIL_ELEM   (NQUAD * 4)                 // 151668, single scalar tail
#define Q_EOS       (EOS_TOK / 4)               // 37911 (EOS at lane 1)

typedef __attribute__((ext_vector_type(4))) unsigned int v4u;
typedef __attribute__((ext_vector_type(8))) int          v8i;
typedef __attribute__((ext_vector_type(4))) int          v4i;
typedef __attribute__((ext_vector_type(4))) float        f4;

// ---------------- threefry2x32 (bit-exact JAX schedule) ----------------
__device__ __forceinline__ void tf2x32(unsigned k0, unsigned k1,
                                       unsigned c0, unsigned c1,
                                       unsigned& o0, unsigned& o1) {
  unsigned ks2 = k0 ^ k1 ^ 0x1BD11BDAu;
  unsigned x0 = c0 + k0, x1 = c1 + k1;
#define TF_R(r) { x0 += x1; x1 = (x1 << (r)) | (x1 >> (32 - (r))); x1 ^= x0; }
#define TF_4A  TF_R(13) TF_R(15) TF_R(26) TF_R(6)
#define TF_4B  TF_R(17) TF_R(29) TF_R(16) TF_R(24)
  TF_4A; x0 += k1;  x1 += ks2 + 1u;
  TF_4B; x0 += ks2; x1 += k0  + 2u;
  TF_4A; x0 += k0;  x1 += k1  + 3u;
  TF_4B; x0 += k1;  x1 += ks2 + 4u;
  TF_4A; x0 += ks2; x1 += k0  + 5u;
#undef TF_R
#undef TF_4A
#undef TF_4B
  o0 = x0; o1 = x1;
}

// random_bits flat element j of an n-element draw (n even), n2 = n/2
__device__ __forceinline__ unsigned jax_bits(unsigned k0, unsigned k1,
                                             unsigned j, unsigned n2) {
  unsigned o0, o1;
  if (j < n2) { tf2x32(k0, k1, j, j + n2, o0, o1); return o0; }
  tf2x32(k0, k1, j - n2, j, o0, o1); return o1;
}

__device__ __forceinline__ float jax_gumbel(unsigned k0, unsigned k1, unsigned v) {
  unsigned bits = jax_bits(k0, k1, v, VOCAB / 2);
  float f = __uint_as_float(0x3F800000u | (bits >> 9)) - 1.0f;  // [0,1)
  float u = fmaxf(TINYF, f * (1.0f - TINYF) + TINYF);
  return -logf(-logf(u));
}

// ---------------- monotonic float<->uint order map ----------------
__device__ __forceinline__ unsigned f2key(float f) {
  unsigned b = __float_as_uint(f);
  return (b & 0x80000000u) ? ~b : (b | 0x80000000u);
}
__device__ __forceinline__ float key2f(unsigned k) {
  unsigned b = (k & 0x80000000u) ? (k & 0x7FFFFFFFu) : ~k;
  return __uint_as_float(b);
}

// ---------------- TDM: 1-D tile, global -> LDS (CDNA5 Tensor Data Mover) ----
// The builtin only exists in the gfx1250 device pass; the host pass merely
// needs to parse a stub (device code is never code-generated for host).
#if defined(__HIP_DEVICE_COMPILE__)
#if __has_builtin(__builtin_amdgcn_tensor_load_to_lds)
__device__ __forceinline__ void tdm_load_1d(uint32_t lds_off, const void* gptr,
                                            uint32_t nelem) {
  uint64_t ga = (uint64_t)(uintptr_t)gptr;
  v4u g0;
  g0.x = 1u;                                   // count=1, no gather, user mode
  g0.y = lds_off;                              // lds_addr (bytes)
  g0.z = (uint32_t)(ga & 0xFFFFFFFFu);         // global_addr[31:0]
  g0.w = (uint32_t)((ga >> 32) & 0x01FFFFFFu)  // global_addr[56:32]
       | (2u << 30);                           // type = 2 ("image")
  v8i g1;
  g1[0] = (int)(2u << 16);                     // wg_mask=0, data_size=2 (4B)
  g1[1] = (int)((nelem & 0xFFFFu) << 16);      // tensor_dim0[15:0]
  g1[2] = (int)(((nelem >> 16) & 0xFFFFu)      // tensor_dim0[31:16]
       | (1u << 16));                          // tensor_dim1 = 1
  g1[3] = (int)((nelem & 0xFFFFu) << 16);      // tile_dim0 = nelem
  g1[4] = 1;                                   // tile_dim1 = 1, tile_dim2 unused
  g1[5] = (int)nelem;                          // tensor_dim0_stride[31:0]
  g1[6] = 0;
  g1[7] = 0;
  v4i z4 = {0, 0, 0, 0};
#if defined(__clang_major__) && (__clang_major__ >= 23)
  v8i z8 = {0, 0, 0, 0, 0, 0, 0, 0};
  __builtin_amdgcn_tensor_load_to_lds(g0, g1, z4, z4, z8, 0);
#else
  __builtin_amdgcn_tensor_load_to_lds(g0, g1, z4, z4, 0);
#endif
}
#else
#error "gfx1250 device pass lacks __builtin_amdgcn_tensor_load_to_lds"
#endif
#else   // host pass: parse-only stub, never executed or code-generated
__device__ __forceinline__ void tdm_load_1d(uint32_t, const void*, uint32_t) {}
#endif

__device__ __forceinline__ void tdm_wait0() {
#if defined(__HIP_DEVICE_COMPILE__)
  __builtin_amdgcn_s_wait_tensorcnt(0);
#endif
}

// ---------------- fused kernel: one block per decode step ----------------
// Phase A: wave0 issues 2 TDM descriptors (candidate slices -> LDS) while all
//          waves stream NT zero stores over the row's masked region.
// Phase B: top-64 selection in LDS, top-p cutoff, threefry/Gumbel token draw,
//          write the 4108 active probabilities.
__global__ __launch_bounds__(NTHREADS)
void sample_rows_kernel(const float* __restrict__ logits, float* __restrict__ out) {
  const int t    = blockIdx.x;
  const int tid  = threadIdx.x;
  const int lane = tid & 31;
  const int wid  = tid >> 5;

  __shared__ float sc_arr[NCAND];     // cond candidates
  __shared__ float su_arr[NCAND];     // uncond candidates
  __shared__ unsigned keys[NCAND];    // mutable order keys
  __shared__ unsigned long long red[NTHREADS / 32];
  __shared__ unsigned skey[TOPK];
  __shared__ int      sidx[TOPK];
  __shared__ float    scoreArr[TOPK];
  __shared__ float    bc[3];          // s0/TEMP, cutoff, 1/Zf
  __shared__ int      shJ;

  const float* cond = logits + (size_t)t * 2u * VOCAB;
  const float* unc  = cond + VOCAB;
  float* prow = out + T_STEPS + (size_t)t * VOCAB;

  // ---- Phase A: kick off DMA, then hide it under the zero-fill ----
  if (wid == 0) {     // single wave issues both DMA descriptors (EXEC-agnostic)
    tdm_load_1d((uint32_t)(uintptr_t)&sc_arr[0], cond + AUDIO_START, NAUDIO);
    tdm_load_1d((uint32_t)(uintptr_t)&su_arr[0], unc  + AUDIO_START, NAUDIO);
  }

  // NT zero-fill of the masked region [0, AUDIO_START), skipping the EOS lane
  // (EOS column is written exactly once, by the prob pass below).
  {
    f4 z = {0.0f, 0.0f, 0.0f, 0.0f};
    f4* p4 = (f4*)prow;
    for (int q = tid; q < NQUAD; q += NTHREADS) {
      if (q == Q_EOS) {               // quad {151644, EOS, 151646, 151647}
        __builtin_nontemporal_store(0.0f, &prow[EOS_TOK - 1]);
        __builtin_nontemporal_store(0.0f, &prow[EOS_TOK + 1]);
        __builtin_nontemporal_store(0.0f, &prow[EOS_TOK + 2]);
      } else {
        __builtin_nontemporal_store(z, p4 + q);
      }
    }
    if (tid == 34) __builtin_nontemporal_store(0.0f, &prow[TAIL_ELEM]);
  }

  if (tid == 32) sc_arr[NCAND - 1] = cond[EOS_TOK];
  if (tid == 33) su_arr[NCAND - 1] = unc [EOS_TOK];
  if (wid == 0) tdm_wait0();
  __syncthreads();

  // ---- cfg = uncond + 2*(cond-uncond); build order keys ----
  for (int c = tid; c < NCAND; c += NTHREADS) {
    float cv = sc_arr[c], uv = su_arr[c];
    float cfg = uv + 2.0f * (cv - uv);
    if (c == NCAND - 1 && t <= 1) cfg = NEGF;   // EOS forbidden for step <= MIN_TOKENS
    keys[c] = f2key(cfg);
  }
  __syncthreads();

  // ---- iterative top-64 selection (sorted descending) ----
  for (int k = 0; k < TOPK; ++k) {
    unsigned long long best = 0ull;
    for (int c = tid; c < NCAND; c += NTHREADS) {
      unsigned long long p =
          ((unsigned long long)keys[c] << 32) | (unsigned)(0xFFFFu - c);
      if (p > best) best = p;
    }
    #pragma unroll
    for (int off = 16; off > 0; off >>= 1) {
      unsigned long long o = __shfl_down(best, off);   // wave32
      if (o > best) best = o;
    }
    if (lane == 0) red[wid] = best;
    __syncthreads();
    if (tid == 0) {
      unsigned long long b = red[0];
      #pragma unroll
      for (int w = 1; w < NTHREADS / 32; ++w)
        if (red[w] > b) b = red[w];
      unsigned key = (unsigned)(b >> 32);
      int c = 0xFFFF - (int)(b & 0xFFFFu);
      skey[k] = key; sidx[k] = c;
      keys[c] = 0u;                     // remove from further rounds
    }
    __syncthreads();
  }

  // ---- top-p cutoff + normalizers (serial over 64, trivial) ----
  if (tid == 0) {
    float s0 = key2f(skey[0]);
    float Z = 0.0f;
    for (int k = 0; k < TOPK; ++k) {
      float e = expf(key2f(skey[k]) - s0);
      scoreArr[k] = e;
      Z += e;
    }
    float cum = 0.0f; int J = 0;
    for (int k = 1; k < TOPK; ++k) {           // keep k iff cum(p[0..k-1]) <= 0.9
      cum += scoreArr[k - 1] / Z;
      if (cum <= TOPP) J = k; else break;
    }
    float cutoff = key2f(skey[J]);
    float s0d = s0 / TEMP;
    float Zf = 0.0f;
    for (int k = 0; k <= J; ++k)
      Zf += expf(key2f(skey[k]) / TEMP - s0d);
    bc[0] = s0d; bc[1] = cutoff; bc[2] = 1.0f / Zf;
    shJ = J;
  }
  __syncthreads();

  // ---- Gumbel draws for surviving candidates ----
  const int J = shJ;
  if (tid < TOPK) {
    float s = -INFINITY;
    if (tid <= J) {
      // split(key(1), 512)[t] : base threefry key (0,1), counters over 1024
      unsigned kt0 = jax_bits(0u, 1u, (unsigned)(2 * t),     512u);
      unsigned kt1 = jax_bits(0u, 1u, (unsigned)(2 * t + 1), 512u);
      int c = sidx[tid];
      unsigned v = (c == NCAND - 1) ? (unsigned)EOS_TOK
                                    : (unsigned)(AUDIO_START + c);
      float g = jax_gumbel(kt0, kt1, v);
      s = key2f(skey[tid]) / TEMP + g;
    }
    scoreArr[tid] = s;
  }
  __syncthreads();
  if (tid == 0) {
    float best = -INFINITY; int bestv = 0x7FFFFFFF;
    for (int k = 0; k <= J; ++k) {
      int c = sidx[k];
      int v = (c == NCAND - 1) ? EOS_TOK : (AUDIO_START + c);
      float s = scoreArr[k];
      if (s > best || (s == best && v < bestv)) { best = s; bestv = v; }
    }
    out[t] = (float)bestv;                      // tokens, flat-concat first
  }

  // ---- write active probabilities (covers [AUDIO_START, V) and EOS) ----
  float s0d = bc[0], cutoff = bc[1], invZf = bc[2];
  for (int c = tid; c < NCAND; c += NTHREADS) {
    float cv = sc_arr[c], uv = su_arr[c];
    float cfg = uv + 2.0f * (cv - uv);
    bool eos_slot = (c == NCAND - 1);
    int v = eos_slot ? EOS_TOK : (AUDIO_START + c);
    float p = 0.0f;
    if (!(eos_slot && t <= 1) && cfg >= cutoff)
      p = expf(cfg / TEMP - s0d) * invZf;
    prow[v] = p;
  }
}

extern "C" void kernel_launch(void* const* d_in, const int* in_sizes, int n_in,
                              void* d_out, int out_size, void* d_ws, size_t ws_size,
                              hipStream_t stream) {
  (void)in_sizes; (void)n_in; (void)d_ws; (void)ws_size; (void)out_size;
  const float* logits = (const float*)d_in[0];
  float* out = (float*)d_out;

  // One block per decode step: TDM-stage candidates into LDS while the block
  // NT-zero-fills its row's masked region, then select/sample/write.
  sample_rows_kernel<<<T_STEPS, NTHREADS, 0, stream>>>(logits, out);
}